// ScatLayerj2_4440996184633
// MI455X (gfx1250) — compile-verified
//
#include <hip/hip_runtime.h>
#include <math.h>

typedef float v2f __attribute__((ext_vector_type(2)));
typedef float v8f __attribute__((ext_vector_type(8)));

#define MAGB2 1.0e-4f
#define MAGB  0.01f

__device__ __forceinline__ int reflect_idx(int i, int n) {
  while (i < 0 || i >= n) {
    if (i < 0) i = -1 - i;
    else       i = 2 * n - 1 - i;
  }
  return i;
}

// band(d, h): h[d] if 0 <= d < L else 0, fully unrolled so taps stay in VGPRs
__device__ __forceinline__ float band5(int d, const float* h) {
  float v = 0.f;
#pragma unroll
  for (int k = 0; k < 5; ++k) if (d == k) v = h[k];
  return v;
}
__device__ __forceinline__ float band7(int d, const float* h) {
  float v = 0.f;
#pragma unroll
  for (int k = 0; k < 7; ++k) if (d == k) v = h[k];
  return v;
}
__device__ __forceinline__ float band10(int d, const float* h) {
  float v = 0.f;
#pragma unroll
  for (int k = 0; k < 10; ++k) if (d == k) v = h[k];
  return v;
}

// Opaque register barrier: prevents LLVM from sinking/rematerializing the
// band-select chains into the WMMA loop.
__device__ __forceinline__ void pin_vgpr(float& x) {
  asm volatile("" : "+v"(x));
}

// ---------------------------------------------------------------------------
// Level-1 (full-rate) DTCWT stage.
// MODE 0: x(8,C=3,512,512) -> ll_out = ll_full (same layout), mag_out = p(8,18,256,256)
// MODE 1: p(8,C=18,256,256) -> ll_out = d_out (pooled ll, ch 3+pc),
//                              mag_out = d_out (mags, ch 39 + o*18 + pc)
// Block: 128 threads (4 waves), tile 32x32 full-res.
// ---------------------------------------------------------------------------
template <int MODE>
__global__ __launch_bounds__(128) void j1_kernel(
    const float* __restrict__ src,
    const float* __restrict__ h0o, const float* __restrict__ h1o,
    float* __restrict__ ll_out, float* __restrict__ mag_out,
    int C, int H, int W)
{
  __shared__ float xt[38][39];    // reflect-padded input tile (halo 3)
  __shared__ float lof[40][33];   // horizontal lowpass  (rows 38,39 zeroed pad)
  __shared__ float hif[40][33];   // horizontal highpass
  __shared__ float tll[32][33];
  __shared__ float tlh[32][33];
  __shared__ float thl[32][33];
  __shared__ float thh[32][33];

  const int tid = threadIdx.x;
  const int bx = blockIdx.x, by = blockIdx.y;
  const int n = blockIdx.z / C, c = blockIdx.z % C;
  const int X0 = bx * 32, Y0 = by * 32;
  const size_t base = ((size_t)(n * C + c)) * H * W;

  float f0[5], f1[7];
#pragma unroll
  for (int k = 0; k < 5; ++k) f0[k] = h0o[k];
#pragma unroll
  for (int k = 0; k < 7; ++k) f1[k] = h1o[k];

  // Phase 1: load tile + reflect halo
  for (int idx = tid; idx < 38 * 38; idx += 128) {
    int r = idx / 38, cc = idx - r * 38;
    xt[r][cc] = src[base + (size_t)reflect_idx(Y0 - 3 + r, H) * W
                         + reflect_idx(X0 - 3 + cc, W)];
  }
  __syncthreads();

  // Phase 2: horizontal pass (lo: h0o len5 m=2; hi: h1o len7 m=3)
  for (int idx = tid; idx < 38 * 32; idx += 128) {
    int r = idx >> 5, j = idx & 31;
    float lo = 0.f, hi = 0.f;
#pragma unroll
    for (int k = 0; k < 5; ++k) lo += f0[k] * xt[r][j + 1 + k];
#pragma unroll
    for (int k = 0; k < 7; ++k) hi += f1[k] * xt[r][j + k];
    lof[r][j] = lo; hif[r][j] = hi;
  }
  // zero the 2 padding rows read by the zero-band WMMA chunks
  for (int idx = tid; idx < 2 * 32; idx += 128) {
    int r = 38 + (idx >> 5), j = idx & 31;
    lof[r][j] = 0.f; hif[r][j] = 0.f;
  }
  __syncthreads();

  // Phase 3: vertical pass as banded matmul via V_WMMA_F32_16X16X4_F32.
  // y[ro+i][j] = sum_t A[i][t] * S[ro+t][j], A[i][t] = h[t-i-shift], K = 24.
  {
    const int wave = tid >> 5, lane = tid & 31;
    const int l15 = lane & 15, half = lane >> 4;
    const float (*S)[33] = (wave < 2) ? lof : hif;
    float (*D)[33] = (wave == 0) ? tll : (wave == 1) ? tlh
                    : (wave == 2) ? thl : thh;
    const bool useH1 = (wave & 1) != 0;

    // Hoist A fragments once per wave, pinned so they can't be rematerialized
    float ax[6], ay[6];
#pragma unroll
    for (int kc = 0; kc < 6; ++kc) {
      const int kb = kc * 4 + half * 2;
      ax[kc] = useH1 ? band7(kb - l15, f1)     : band5(kb - l15 - 1, f0);
      ay[kc] = useH1 ? band7(kb + 1 - l15, f1) : band5(kb - l15, f0);
      pin_vgpr(ax[kc]); pin_vgpr(ay[kc]);
    }

#pragma unroll
    for (int ro = 0; ro < 32; ro += 16) {
#pragma unroll
      for (int co = 0; co < 32; co += 16) {
        v8f acc = {0.f, 0.f, 0.f, 0.f, 0.f, 0.f, 0.f, 0.f};
#pragma unroll
        for (int kc = 0; kc < 6; ++kc) {
          const int kb = kc * 4 + half * 2;   // K index pair handled by this lane half
          v2f a, b;
          a.x = ax[kc]; a.y = ay[kc];
          b.x = S[ro + kb][co + l15];
          b.y = S[ro + kb + 1][co + l15];
          acc = __builtin_amdgcn_wmma_f32_16x16x4_f32(
              false, a, false, b, (short)0, acc, false, false);
        }
#pragma unroll
        for (int v = 0; v < 8; ++v)
          D[ro + half * 8 + v][co + l15] = acc[v];
      }
    }
  }
  __syncthreads();

  // Phase 4: q2c + magnitude
  for (int idx = tid; idx < 6 * 256; idx += 128) {
    const int o = idx >> 8, p2 = idx & 255;
    const int y = p2 >> 4, x2 = p2 & 15;
    const float (*Sarr)[33] = (o == 0 || o == 5) ? tlh
                            : (o == 1 || o == 4) ? thh : thl;
    const float av = Sarr[2 * y][2 * x2],     bv = Sarr[2 * y][2 * x2 + 1];
    const float cv = Sarr[2 * y + 1][2 * x2], dv = Sarr[2 * y + 1][2 * x2 + 1];
    float re, im;
    if (o <= 2) { re = 0.5f * (av + dv); im = 0.5f * (bv - cv); }
    else        { re = 0.5f * (av - dv); im = 0.5f * (bv + cv); }
    const float m = sqrtf(re * re + im * im + MAGB2) - MAGB;
    const int yh = (Y0 >> 1) + y, xh = (X0 >> 1) + x2;
    if (MODE == 0) {
      mag_out[(((size_t)n * (6 * C) + o * C + c) * (H >> 1) + yh) * (W >> 1) + xh] = m;
    } else {
      mag_out[(((size_t)n * 147 + 39 + o * 18 + c) * (H >> 1) + yh) * (W >> 1) + xh] = m;
    }
  }
  if (MODE == 0) {
    for (int idx = tid; idx < 32 * 32; idx += 128) {
      int y = idx >> 5, x2 = idx & 31;
      ll_out[base + (size_t)(Y0 + y) * W + (X0 + x2)] = tll[y][x2];
    }
  } else {
    for (int idx = tid; idx < 256; idx += 128) {
      int y = idx >> 4, x2 = idx & 15;
      float m = 0.25f * (tll[2*y][2*x2] + tll[2*y][2*x2+1] +
                         tll[2*y+1][2*x2] + tll[2*y+1][2*x2+1]);
      ll_out[(((size_t)n * 147 + 3 + c) * (H >> 1) + (Y0 >> 1) + y) * (W >> 1)
             + (X0 >> 1) + x2] = m;
    }
  }
}

// ---------------------------------------------------------------------------
// Level-2 (stride-2 dual-phase) stage on ll_full(8,3,512,512).
// Block: 256 threads (8 waves), 16x16 output tile at 128-res.
// Vertical polyphase filters as stride-2 band matmuls: ya[t]=sum_k h[k]e[2t+k],
// A[t][u]=h[u-2t], K=40 -> 10 WMMA chunks per 16x16 tile.
// ---------------------------------------------------------------------------
__global__ __launch_bounds__(256) void j2_kernel(
    const float* __restrict__ src,
    const float* __restrict__ h0a, const float* __restrict__ h0b,
    const float* __restrict__ h1a, const float* __restrict__ h1b,
    float* __restrict__ out)
{
  __shared__ float xt[80][81];   // reflect-padded 512-res tile
  __shared__ float lof[80][33];  // horizontal dual-phase lowpass (256-res cols)
  __shared__ float hif[80][33];
  __shared__ float tll[32][33];  // 256-res vertical outputs
  __shared__ float tlh[32][33];
  __shared__ float thl[32][33];
  __shared__ float thh[32][33];

  const int tid = threadIdx.x;
  const int bx = blockIdx.x, by = blockIdx.y;
  const int n = blockIdx.z / 3, c = blockIdx.z % 3;
  const int R0 = by * 64 - 8, C0 = bx * 64 - 8;
  const size_t base = ((size_t)(n * 3 + c)) * 512 * 512;

  float f0a[10], f0b[10], f1a[10], f1b[10];
#pragma unroll
  for (int k = 0; k < 10; ++k) {
    f0a[k] = h0a[k]; f0b[k] = h0b[k]; f1a[k] = h1a[k]; f1b[k] = h1b[k];
  }

  // Phase 1: 80x80 reflect-padded tile of ll_full
  for (int idx = tid; idx < 80 * 80; idx += 256) {
    int r = idx / 80, cc = idx - r * 80;
    xt[r][cc] = src[base + (size_t)reflect_idx(R0 + r, 512) * 512
                         + reflect_idx(C0 + cc, 512)];
  }
  __syncthreads();

  // Phase 2: horizontal dual-phase (stride-2) pass -> 32 cols at 256-res
  for (int idx = tid; idx < 80 * 32; idx += 256) {
    int r = idx >> 5, j = idx & 31;
    float lo = 0.f, hi = 0.f;
    if ((j & 1) == 0) {        // even 256-res col: lo<-h0a(even taps), hi<-h1b(odd taps)
#pragma unroll
      for (int k = 0; k < 10; ++k) {
        lo += f0a[k] * xt[r][2 * j + 2 * k];
        hi += f1b[k] * xt[r][2 * j + 2 * k + 1];
      }
    } else {                   // odd col: lo<-h0b(odd taps), hi<-h1a(even taps)
#pragma unroll
      for (int k = 0; k < 10; ++k) {
        lo += f0b[k] * xt[r][2 * j + 2 * k - 1];
        hi += f1a[k] * xt[r][2 * j + 2 * k - 2];
      }
    }
    lof[r][j] = lo; hif[r][j] = hi;
  }
  __syncthreads();

  // Phase 3: vertical polyphase via WMMA. Wave w: src=(w<4?lo:hi), fs=w&3:
  // fs0:h0a/even src rows -> dest rows 2t   (ll|hl)
  // fs1:h0b/odd            -> dest rows 2t+1 (ll|hl)
  // fs2:h1a/even           -> dest rows 2t+1 (lh|hh)   (highpass swap)
  // fs3:h1b/odd            -> dest rows 2t   (lh|hh)
  {
    const int w = tid >> 5, lane = tid & 31;
    const int l15 = lane & 15, half = lane >> 4;
    const int fs = w & 3;
    const float (*S)[33] = (w < 4) ? lof : hif;
    float (*D)[33] = (w < 4) ? ((fs < 2) ? tll : tlh)
                             : ((fs < 2) ? thl : thh);
    const int sodd = (fs == 1 || fs == 3) ? 1 : 0;
    const int rodd = (fs == 1 || fs == 2) ? 1 : 0;
    float fk[10];
#pragma unroll
    for (int k = 0; k < 10; ++k)
      fk[k] = (fs == 0) ? f0a[k] : (fs == 1) ? f0b[k]
            : (fs == 2) ? f1a[k] : f1b[k];

    // Hoist A fragments once per wave, pinned against rematerialization
    float ax[10], ay[10];
#pragma unroll
    for (int kc = 0; kc < 10; ++kc) {
      const int kb = kc * 4 + half * 2;
      ax[kc] = band10(kb - 2 * l15, fk);
      ay[kc] = band10(kb + 1 - 2 * l15, fk);
      pin_vgpr(ax[kc]); pin_vgpr(ay[kc]);
    }

#pragma unroll
    for (int co = 0; co < 32; co += 16) {
      v8f acc = {0.f, 0.f, 0.f, 0.f, 0.f, 0.f, 0.f, 0.f};
#pragma unroll
      for (int kc = 0; kc < 10; ++kc) {
        const int kb = kc * 4 + half * 2;
        v2f a, b;
        a.x = ax[kc]; a.y = ay[kc];
        b.x = S[2 * kb + sodd][co + l15];
        b.y = S[2 * kb + 2 + sodd][co + l15];
        acc = __builtin_amdgcn_wmma_f32_16x16x4_f32(
            false, a, false, b, (short)0, acc, false, false);
      }
#pragma unroll
      for (int v = 0; v < 8; ++v)
        D[2 * (half * 8 + v) + rodd][co + l15] = acc[v];
    }
  }
  __syncthreads();

  // Phase 4: q2c + magnitude -> out ch 21 + o*3 + c; pooled ll -> out ch c
  const int YH0 = by * 16, XH0 = bx * 16;
  for (int idx = tid; idx < 6 * 256; idx += 256) {
    const int o = idx >> 8, p2 = idx & 255;
    const int y = p2 >> 4, x2 = p2 & 15;
    const float (*Sarr)[33] = (o == 0 || o == 5) ? tlh
                            : (o == 1 || o == 4) ? thh : thl;
    const float av = Sarr[2 * y][2 * x2],     bv = Sarr[2 * y][2 * x2 + 1];
    const float cv = Sarr[2 * y + 1][2 * x2], dv = Sarr[2 * y + 1][2 * x2 + 1];
    float re, im;
    if (o <= 2) { re = 0.5f * (av + dv); im = 0.5f * (bv - cv); }
    else        { re = 0.5f * (av - dv); im = 0.5f * (bv + cv); }
    const float m = sqrtf(re * re + im * im + MAGB2) - MAGB;
    out[(((size_t)n * 147 + 21 + o * 3 + c) * 128 + (YH0 + y)) * 128 + (XH0 + x2)] = m;
  }
  for (int idx = tid; idx < 256; idx += 256) {
    int y = idx >> 4, x2 = idx & 15;
    float m = 0.25f * (tll[2*y][2*x2] + tll[2*y][2*x2+1] +
                       tll[2*y+1][2*x2] + tll[2*y+1][2*x2+1]);
    out[(((size_t)n * 147 + c) * 128 + (YH0 + y)) * 128 + (XH0 + x2)] = m;
  }
}

// ---------------------------------------------------------------------------
extern "C" void kernel_launch(void* const* d_in, const int* in_sizes, int n_in,
                              void* d_out, int out_size, void* d_ws, size_t ws_size,
                              hipStream_t stream) {
  (void)in_sizes; (void)n_in; (void)out_size; (void)ws_size;
  const float* x   = (const float*)d_in[0];
  const float* h0o = (const float*)d_in[1];
  const float* h1o = (const float*)d_in[2];
  const float* h0a = (const float*)d_in[3];
  const float* h0b = (const float*)d_in[4];
  const float* h1a = (const float*)d_in[5];
  const float* h1b = (const float*)d_in[6];
  float* out = (float*)d_out;

  // workspace: ll_full (8,3,512,512) then p (8,18,256,256) = 63 MB total
  float* llf = (float*)d_ws;
  float* p   = llf + (size_t)8 * 3 * 512 * 512;

  // Stage A: level-1 on x -> ll_full + p
  j1_kernel<0><<<dim3(16, 16, 24), dim3(128), 0, stream>>>(
      x, h0o, h1o, llf, p, 3, 512, 512);

  // Stage B: level-2 on ll_full -> out ch 0..2 (s0), 21..38 (s1_j2)
  j2_kernel<<<dim3(8, 8, 24), dim3(256), 0, stream>>>(
      llf, h0a, h0b, h1a, h1b, out);

  // Stage C: level-1 on p -> out ch 3..20 (pooled s1_j1), 39..146 (s2)
  j1_kernel<1><<<dim3(8, 8, 144), dim3(128), 0, stream>>>(
      p, h0o, h1o, out, out, 18, 256, 256);
}